// SpatialPoolAgent_34411277976194
// MI455X (gfx1250) — compile-verified
//
#include <hip/hip_runtime.h>
#include <hip/hip_bf16.h>
#include <cstdint>

// ---------------------------------------------------------------------------
// SpatialPoolAgent on MI455X (gfx1250)
//   out = input_grid;  out[b,c,0,0] = max(in[b,c,0,0], segmax_b(enc[:,c]))
// Memory-bound: ~272MB traffic -> ~12us at 23.3 TB/s. No matrix math exists
// in this op (max-reduction + copy), so the CDNA5-specific path used is the
// async global<->LDS DMA (ASYNCcnt) pipeline for the 256MB grid copy, plus
// wave32-shaped reductions (lane == channel, C == 32).
// ---------------------------------------------------------------------------

#define AS1 __attribute__((address_space(1)))
#define AS3 __attribute__((address_space(3)))

typedef int v4i __attribute__((ext_vector_type(4)));

#if defined(__gfx1250__) && \
    __has_builtin(__builtin_amdgcn_global_load_async_to_lds_b128) && \
    __has_builtin(__builtin_amdgcn_global_store_async_from_lds_b128) && \
    __has_builtin(__builtin_amdgcn_s_wait_asynccnt)
#define USE_GFX1250_ASYNC 1
#else
#define USE_GFX1250_ASYNC 0
#endif

#define COPY_TPB   256
#define COPY_ITEMS 4   // float4 per thread per block-tile (16KB LDS staging)

#define NEG_INF (-__builtin_huge_valf())

// ---------------------------------------------------------------------------
// Stage 1: per (scene, slice) partial max over agents. C == 32 == wave32:
// lane c of each wave reads channel c; one 128B row per wave per agent.
// ---------------------------------------------------------------------------
__global__ void seg_partial_kernel(const float* __restrict__ enc,
                                   const int* __restrict__ nagents,
                                   float* __restrict__ partial,
                                   int split, long long K) {
    const int blk = blockIdx.x;
    const int b = blk / split;
    const int j = blk % split;

    // Scene boundaries from running sum of num_agents (uniform scalar loop).
    long long start = 0;
    for (int i = 0; i < b; ++i) start += (long long)nagents[i];
    long long nb = (long long)nagents[b];
    long long end = start + nb;
    if (start > K) start = K;
    if (end > K) end = K;
    nb = end - start;

    const long long chunk = (nb + split - 1) / split;
    long long s = start + (long long)j * chunk;
    long long e = s + chunk;
    if (s > end) s = end;
    if (e > end) e = end;

    const int lane = threadIdx.x & 31;   // channel
    const int w    = threadIdx.x >> 5;   // wave id (8 waves)

    float m = NEG_INF;
    for (long long k = s + w; k < e; k += 8) {
        m = fmaxf(m, enc[k * 32 + lane]);
    }

    __shared__ float red[8][32];
    red[w][lane] = m;
    __syncthreads();
    if (w == 0) {
        float r = red[0][lane];
#pragma unroll
        for (int i = 1; i < 8; ++i) r = fmaxf(r, red[i][lane]);
        partial[(long long)blk * 32 + lane] = r;
    }
}

// ---------------------------------------------------------------------------
// Stage 2: fold the `split` partials per scene. One wave32 per scene.
// ---------------------------------------------------------------------------
__global__ void seg_combine_kernel(const float* __restrict__ partial,
                                   float* __restrict__ seg, int split) {
    const int b = blockIdx.x;
    const int lane = threadIdx.x;   // 32 threads
    float r = NEG_INF;
    for (int j = 0; j < split; ++j)
        r = fmaxf(r, partial[((long long)b * split + j) * 32 + lane]);
    seg[(long long)b * 32 + lane] = r;
}

// ---------------------------------------------------------------------------
// Bulk grid copy (128MB -> 128MB). On gfx1250: async B128 DMA through LDS,
// no VGPR round trip; one s_wait_asynccnt 0 between load and store batches.
// S_ENDPGM's implicit wait-idle covers outstanding async stores.
// ---------------------------------------------------------------------------
__global__ void grid_copy_kernel(const float4* __restrict__ src,
                                 float4* __restrict__ dst, long long n4) {
    const long long base =
        (long long)blockIdx.x * (COPY_TPB * COPY_ITEMS) + threadIdx.x;
#if USE_GFX1250_ASYNC
    __shared__ float4 tile[COPY_TPB * COPY_ITEMS];
    const int slot = threadIdx.x;
#pragma unroll
    for (int i = 0; i < COPY_ITEMS; ++i) {
        long long idx = base + (long long)i * COPY_TPB;
        if (idx < n4) {
            __builtin_amdgcn_global_load_async_to_lds_b128(
                (AS1 v4i*)(uintptr_t)(src + idx),
                (AS3 v4i*)(unsigned)(uintptr_t)(tile + i * COPY_TPB + slot),
                0, 0);
        }
    }
    __builtin_amdgcn_s_wait_asynccnt(0);
#pragma unroll
    for (int i = 0; i < COPY_ITEMS; ++i) {
        long long idx = base + (long long)i * COPY_TPB;
        if (idx < n4) {
            __builtin_amdgcn_global_store_async_from_lds_b128(
                (AS1 v4i*)(uintptr_t)(dst + idx),
                (AS3 v4i*)(unsigned)(uintptr_t)(tile + i * COPY_TPB + slot),
                0, 0);
        }
    }
#else
#pragma unroll
    for (int i = 0; i < COPY_ITEMS; ++i) {
        long long idx = base + (long long)i * COPY_TPB;
        if (idx < n4) dst[idx] = src[idx];
    }
#endif
}

// ---------------------------------------------------------------------------
// Patch the (0,0) cell of each [b,c] plane: out = max(in, seg_max).
// Runs after grid_copy (stream-ordered), so the overwrite is safe.
// ---------------------------------------------------------------------------
__global__ void fixup_kernel(const float* __restrict__ grid_in,
                             const float* __restrict__ seg,
                             float* __restrict__ out,
                             int nplanes, long long hw) {
    const int p = blockIdx.x * blockDim.x + threadIdx.x;
    if (p < nplanes) {
        const long long off = (long long)p * hw;
        out[off] = fmaxf(grid_in[off], seg[p]);
    }
}

// ---------------------------------------------------------------------------
extern "C" void kernel_launch(void* const* d_in, const int* in_sizes, int n_in,
                              void* d_out, int out_size, void* d_ws, size_t ws_size,
                              hipStream_t stream) {
    const float* grid_in = (const float*)d_in[0];   // [B,C,H,W] f32
    const float* enc     = (const float*)d_in[1];   // [K,C]     f32
    // d_in[2]: encode_coordinates [K,2] — unused by the reference math.
    const int* nagents   = (const int*)d_in[3];     // [B] int32 (JAX-canonicalized)
    float* out = (float*)d_out;

    const long long K  = (long long)in_sizes[2] / 2;
    const int       C  = (int)((long long)in_sizes[1] / K);          // 32
    const int       B  = in_sizes[3];                                 // 64
    const long long HW = (long long)in_sizes[0] / ((long long)B * C); // 16384

    // Workspace layout: [seg: B*C floats][partial: B*split*C floats]
    int split = 16;
    size_t need = (size_t)B * C * sizeof(float) * (size_t)(1 + split);
    if (ws_size < need) split = 1;
    float* seg     = (float*)d_ws;
    float* partial = seg + (size_t)B * C;

    seg_partial_kernel<<<B * split, 256, 0, stream>>>(enc, nagents, partial, split, K);
    seg_combine_kernel<<<B, 32, 0, stream>>>(partial, seg, split);

    const long long n4 = (long long)out_size / 4;
    const int nblocks =
        (int)((n4 + (long long)COPY_TPB * COPY_ITEMS - 1) / ((long long)COPY_TPB * COPY_ITEMS));
    grid_copy_kernel<<<nblocks, COPY_TPB, 0, stream>>>(
        (const float4*)grid_in, (float4*)out, n4);

    const int nplanes = B * C;
    fixup_kernel<<<(nplanes + 255) / 256, 256, 0, stream>>>(grid_in, seg, out, nplanes, HW);
}